// MAPLoss_Pretrain_87454124081565
// MI455X (gfx1250) — compile-verified
//
#include <hip/hip_runtime.h>
#include <math.h>
#include <stdint.h>

#define BATCH   8
#define HWPIX   (512 * 512)          // 2^18 pixels per image
#define NPIX    (BATCH * HWPIX)      // 2^21 pixels total
#define TILE    512                  // pixels per block iteration
#define BLOCK   128                  // 4 wave32 waves per block
#define PPT     4                    // pixels per thread (TILE/BLOCK)
#define GRID    2048
#define NTILES  (NPIX / TILE)        // 4096  -> 2 tiles per block
#define CHUNKS  ((TILE * 9) / 4)     // 1152 b128 chunks per tile
#define CPT     (CHUNKS / BLOCK)     // 9 async DMA ops per thread (uniform!)

// ---------------- wave32 reductions -----------------------------------------
#if __has_builtin(__builtin_amdgcn_wmma_f32_16x16x4_f32)
typedef __attribute__((ext_vector_type(2))) float v2f;
typedef __attribute__((ext_vector_type(8))) float v8f;
// Sum 32 lane values with one V_WMMA_F32_16X16X4_F32 against an all-ones B.
// A (16x4): lane m -> A[m,0] (lanes 0-15) / A[m,2] (lanes 16-31), other K = 0.
// D[m,n] = x_m + x_{m+16}; per-lane sum of the 8 D VGPRs gives the half-sums,
// one shfl_xor(16) completes the 32-lane sum. EXEC is all-ones here.
__device__ __forceinline__ float wave_sum32(float x) {
    v2f a; a[0] = x;    a[1] = 0.0f;
    v2f b; b[0] = 1.0f; b[1] = 1.0f;
    v8f c = {};
    c = __builtin_amdgcn_wmma_f32_16x16x4_f32(
            false, a, false, b, (short)0, c, false, false);
    float s = ((c[0] + c[1]) + (c[2] + c[3])) + ((c[4] + c[5]) + (c[6] + c[7]));
    s += __shfl_xor(s, 16, 32);
    return s;
}
#else
__device__ __forceinline__ float wave_sum32(float x) {
    #pragma unroll
    for (int o = 16; o > 0; o >>= 1) x += __shfl_xor(x, o, 32);
    return x;
}
#endif

__device__ __forceinline__ float wave_max32(float x) {
    #pragma unroll
    for (int o = 16; o > 0; o >>= 1) x = fmaxf(x, __shfl_xor(x, o, 32));
    return x;
}

// ---------------- pass 1: per-block partial (sum, max) ----------------------
__global__ __launch_bounds__(BLOCK) void map_loss_partial(
    const float* __restrict__ target,
    const float* __restrict__ mu,
    const float* __restrict__ sigY,
    float* __restrict__ ws_sum,
    float* __restrict__ ws_max)
{
    // double-buffered sigma_y tiles: 2 x 18432 B
    __shared__ __align__(16) float s_sig[2][TILE * 9];
    __shared__ float s_ws[BLOCK / 32];
    __shared__ float s_wm[BLOCK / 32];

    const int tid = threadIdx.x;
    float acc = 0.0f;
    float mx  = -__builtin_inff();

    // Issue the 9 async b128 DMAs for one tile into one LDS buffer.
    // gbase is wave-uniform (SGPR pair); per-lane 32-bit voffset; LDS byte
    // address = low 32 bits of the generic shared pointer (flat-aperture rule).
    auto issue_dma = [&](int tile, int bufIdx) {
        const uint64_t gbase =
            (uint64_t)(uintptr_t)(sigY + (size_t)9 * (size_t)tile * TILE);
        #pragma unroll
        for (int j = 0; j < CPT; ++j) {
            const int i = tid + j * BLOCK;
            const uint32_t lds_off =
                (uint32_t)(uintptr_t)(&s_sig[bufIdx][4 * i]);
            const uint32_t voff = (uint32_t)i * 16u;
            asm volatile("global_load_async_to_lds_b128 %0, %1, %2"
                         :
                         : "v"(lds_off), "v"(voff), "s"(gbase)
                         : "memory");
        }
    };

    int buf = 0;
    issue_dma(blockIdx.x, 0);                 // prologue: tile 0 -> buf 0

    for (int tile = blockIdx.x; tile < NTILES; tile += GRID) {
        const int nxt = tile + GRID;
        if (nxt < NTILES) issue_dma(nxt, buf ^ 1);   // pipeline next tile

        // ---- overlap: residual loads (b128) while the DMA is in flight ----
        const int p0 = tile * TILE;                  // tile within one image
        const int b  = p0 >> 18;                     // wave-uniform image idx
        const int hw = (p0 & (HWPIX - 1)) + PPT * tid;
        const size_t base = (size_t)(b * 3) * (size_t)HWPIX + (size_t)hw;
        const float4 t0 = *reinterpret_cast<const float4*>(target + base);
        const float4 t1 = *reinterpret_cast<const float4*>(target + base + HWPIX);
        const float4 t2 = *reinterpret_cast<const float4*>(target + base + 2 * HWPIX);
        const float4 m0 = *reinterpret_cast<const float4*>(mu + base);
        const float4 m1 = *reinterpret_cast<const float4*>(mu + base + HWPIX);
        const float4 m2 = *reinterpret_cast<const float4*>(mu + base + 2 * HWPIX);

        if (nxt < NTILES) {
            // prefetch target at base + 4*HWPIX (image b+1, channel 1)
            __builtin_prefetch(target + base + (size_t)(GRID * TILE), 0, 3);
            // in-order ASYNCcnt: <=9 outstanding => this tile's 9 are done
            asm volatile("s_wait_asynccnt 9" ::: "memory");
        } else {
            asm volatile("s_wait_asynccnt 0" ::: "memory");
        }
        __syncthreads();

        const float dx[PPT] = {t0.x - m0.x, t0.y - m0.y, t0.z - m0.z, t0.w - m0.w};
        const float dy[PPT] = {t1.x - m1.x, t1.y - m1.y, t1.z - m1.z, t1.w - m1.w};
        const float dz[PPT] = {t2.x - m2.x, t2.y - m2.y, t2.z - m2.z, t2.w - m2.w};

        // ---- per-pixel 3x3 adjugate / det / quadratic form (4 pixels) ----
        #pragma unroll
        for (int k = 0; k < PPT; ++k) {
            const float* s = &s_sig[buf][(PPT * tid + k) * 9];
            const float s00 = s[0], s01 = s[1], s02 = s[2];
            const float s10 = s[3], s11 = s[4], s12 = s[5];
            const float s20 = s[6], s21 = s[7], s22 = s[8];

            const float a00 = s11 * s22 - s12 * s21;
            const float a01 = s02 * s21 - s01 * s22;
            const float a02 = s01 * s12 - s02 * s11;
            const float a10 = s12 * s20 - s10 * s22;
            const float a11 = s00 * s22 - s02 * s20;
            const float a12 = s02 * s10 - s00 * s12;
            const float a20 = s10 * s21 - s11 * s20;
            const float a21 = s01 * s20 - s00 * s21;
            const float a22 = s00 * s11 - s01 * s10;
            const float det = s00 * a00 + s01 * a10 + s02 * a20;

            const float d0 = dx[k], d1 = dy[k], d2 = dz[k];
            const float q =
                d0 * (a00 * d0 + a01 * d1 + a02 * d2) +
                d1 * (a10 * d0 + a11 * d1 + a12 * d2) +
                d2 * (a20 * d0 + a21 * d1 + a22 * d2);

            const float t1v = 0.5f * q / det;
            const float t2v = 0.5f * logf(det);
            acc += t1v + t2v;
            mx   = fmaxf(mx, t1v);
        }

        __syncthreads();   // all reads of this buffer done before its reuse
        buf ^= 1;
    }

    // ---- block reduction: WMMA row-sum per wave, then 4-wave combine ----
    const float wsum = wave_sum32(acc);
    const float wmax = wave_max32(mx);
    const int wave = tid >> 5;
    const int lane = tid & 31;
    if (lane == 0) { s_ws[wave] = wsum; s_wm[wave] = wmax; }
    __syncthreads();
    if (tid == 0) {
        float S = 0.0f;
        float M = -__builtin_inff();
        #pragma unroll
        for (int w = 0; w < BLOCK / 32; ++w) {
            S += s_ws[w];
            M  = fmaxf(M, s_wm[w]);
        }
        ws_sum[blockIdx.x] = S;
        ws_max[blockIdx.x] = M;
    }
}

// ---------------- pass 2: fold GRID partials into the scalar ----------------
__global__ __launch_bounds__(256) void map_loss_final(
    const float* __restrict__ ws_sum,
    const float* __restrict__ ws_max,
    float* __restrict__ out)
{
    __shared__ double sd[256];
    __shared__ float  sm[256];
    const int tid = threadIdx.x;

    double a = 0.0;
    float  m = -__builtin_inff();
    for (int i = tid; i < GRID; i += 256) {
        a += (double)ws_sum[i];
        m  = fmaxf(m, ws_max[i]);
    }
    sd[tid] = a;
    sm[tid] = m;
    __syncthreads();
    #pragma unroll
    for (int o = 128; o > 0; o >>= 1) {
        if (tid < o) {
            sd[tid] += sd[tid + o];
            sm[tid]  = fmaxf(sm[tid], sm[tid + o]);
        }
        __syncthreads();
    }
    if (tid == 0) {
        float loss = (float)(sd[0] / (double)NPIX);
        if (sm[0] > 1e8f) loss = 0.0f;   // reference's blow-up guard
        out[0] = loss;
    }
}

extern "C" void kernel_launch(void* const* d_in, const int* in_sizes, int n_in,
                              void* d_out, int out_size, void* d_ws, size_t ws_size,
                              hipStream_t stream) {
    // setup_inputs order: target, mu, sigma_mu (unused), sigma_n (unused), sigma_y
    const float* target = (const float*)d_in[0];
    const float* mu     = (const float*)d_in[1];
    const float* sigY   = (const float*)d_in[4];
    float* out = (float*)d_out;

    float* ws_sum = (float*)d_ws;         // GRID floats
    float* ws_max = ws_sum + GRID;        // GRID floats (16 KB total)

    map_loss_partial<<<GRID, BLOCK, 0, stream>>>(target, mu, sigY, ws_sum, ws_max);
    map_loss_final<<<1, 256, 0, stream>>>(ws_sum, ws_max, out);
    (void)in_sizes; (void)n_in; (void)out_size; (void)ws_size;
}